// Memory_47751446397558
// MI455X (gfx1250) — compile-verified
//
#include <hip/hip_runtime.h>
#include <hip/hip_bf16.h>

#define H_DIM 768
#define T_DIM 2048
#define S_SEG 16                 // parallel segments over T
#define LSEG  (T_DIM / S_SEG)    // 128 timesteps per segment
#define CHUNK 32                 // timesteps staged per LDS buffer
#define NCH   (LSEG / CHUNK)     // 4 chunks per segment
#define HB    64                 // h-channels per block (2 waves of 32)
#define NBUF  2                  // LDS double buffer (1 chunk prefetched)
#define LOG2E 1.44269504088896340736f

// bare v_exp_f32 (exp2)
#define EXP2F(x) __builtin_amdgcn_exp2f(x)

// ---- async global->LDS chunk loader (CDNA5 ASYNCcnt path) -----------------
// Per chunk: 32 rows (t) x 64 floats (h) = 8KB per tensor. 64 threads cover
// 4 rows x 16 b128-columns per issue -> 8 issues x 2 tensors = 16 async
// instructions per thread per chunk (max 32 outstanding <= 63 counter cap).
__device__ __forceinline__ void issue_chunk(const float* kbase, const float* vbase,
                                            unsigned lk0, unsigned lv0,
                                            int c, int tid) {
    const int      row0 = tid >> 4;
    const unsigned colb = (unsigned)(tid & 15) * 16u;
    const int      t0   = c * CHUNK;
#pragma unroll
    for (int i = 0; i < CHUNK / 4; ++i) {
        const int      row  = row0 + i * 4;
        const unsigned goff = (unsigned)((t0 + row) * H_DIM) * 4u + colb;
        const unsigned lofs = (unsigned)(row * HB) * 4u + colb;
        asm volatile("global_load_async_to_lds_b128 %0, %1, %2"
                     :: "v"(lk0 + lofs), "v"(goff), "s"(kbase) : "memory");
        asm volatile("global_load_async_to_lds_b128 %0, %1, %2"
                     :: "v"(lv0 + lofs), "v"(goff), "s"(vbase) : "memory");
    }
}

// ---- Phase 1: per-segment state summary (A, B, P) from zero init ----------
// State exponents are kept in log2 domain: e^x == exp2(x * log2e).
__global__ __launch_bounds__(HB) void wkv_phase1(
    const float* __restrict__ key, const float* __restrict__ value,
    const float* __restrict__ time_decay,
    float* __restrict__ sumA, float* __restrict__ sumB, float* __restrict__ sumP)
{
    __shared__ float lds_k[NBUF][CHUNK * HB];
    __shared__ float lds_v[NBUF][CHUNK * HB];

    const int tid = threadIdx.x;
    const int h0  = blockIdx.x * HB;
    const int j   = blockIdx.y;          // segment (0 .. S_SEG-2)
    const int b   = blockIdx.z;          // batch
    const int h   = h0 + tid;

    const size_t segbase = ((size_t)b * T_DIM + (size_t)j * LSEG) * H_DIM + h0;
    const float* kbase = key   + segbase;
    const float* vbase = value + segbase;

    const float w2 = -__expf(time_decay[h]) * LOG2E;

    const unsigned lk0 = (unsigned)(size_t)&lds_k[0][0];
    const unsigned lk1 = (unsigned)(size_t)&lds_k[1][0];
    const unsigned lv0 = (unsigned)(size_t)&lds_v[0][0];
    const unsigned lv1 = (unsigned)(size_t)&lds_v[1][0];

    issue_chunk(kbase, vbase, lk0, lv0, 0, tid);
    issue_chunk(kbase, vbase, lk1, lv1, 1, tid);

    float aa = 0.0f, bb = 0.0f, pp = -__builtin_inff();

    for (int c = 0; c < NCH; ++c) {
        if (c + 1 < NCH) asm volatile("s_wait_asynccnt 16" ::: "memory");
        else             asm volatile("s_wait_asynccnt 0"  ::: "memory");
        __syncthreads();

        const float* pk = &lds_k[c & 1][tid];
        const float* pv = &lds_v[c & 1][tid];
#pragma unroll 4
        for (int tl = 0; tl < CHUNK; ++tl) {
            const float k2 = pk[tl * HB] * LOG2E;
            const float vt = pv[tl * HB];
            const float pw = pp + w2;
            const float me = fmaxf(pw, k2);
            const float s1 = EXP2F(pw - me);
            const float s2 = EXP2F(k2 - me);
            aa = s1 * aa + s2 * vt;
            bb = s1 * bb + s2;
            pp = me;
        }
        __syncthreads();
        if (c + 2 < NCH)
            issue_chunk(kbase, vbase, (c & 1) ? lk1 : lk0, (c & 1) ? lv1 : lv0,
                        c + 2, tid);
    }

    const size_t o = ((size_t)b * S_SEG + j) * H_DIM + h;
    sumA[o] = aa; sumB[o] = bb; sumP[o] = pp;   // P in log2 domain
}

// ---- Phase 2+3 fused: build prefix state, then replay segment -------------
__global__ __launch_bounds__(HB) void wkv_phase3(
    const float* __restrict__ key, const float* __restrict__ value,
    const float* __restrict__ time_decay, const float* __restrict__ time_first,
    const float* __restrict__ sumA, const float* __restrict__ sumB,
    const float* __restrict__ sumP,
    float* __restrict__ out)
{
    __shared__ float lds_k[NBUF][CHUNK * HB];
    __shared__ float lds_v[NBUF][CHUNK * HB];

    const int tid = threadIdx.x;
    const int h0  = blockIdx.x * HB;
    const int j   = blockIdx.y;
    const int b   = blockIdx.z;
    const int h   = h0 + tid;

    const size_t segbase = ((size_t)b * T_DIM + (size_t)j * LSEG) * H_DIM + h0;
    const float* kbase = key   + segbase;
    const float* vbase = value + segbase;
    float*       obase = out   + segbase + tid;

    const float u2  = time_first[h] * LOG2E;
    const float w2  = -__expf(time_decay[h]) * LOG2E;
    const float Lw2 = (float)LSEG * w2;

    const unsigned lk0 = (unsigned)(size_t)&lds_k[0][0];
    const unsigned lk1 = (unsigned)(size_t)&lds_k[1][0];
    const unsigned lv0 = (unsigned)(size_t)&lds_v[0][0];
    const unsigned lv1 = (unsigned)(size_t)&lds_v[1][0];

    // kick off the fetch pipeline first, then fold the prefix while it flies
    issue_chunk(kbase, vbase, lk0, lv0, 0, tid);
    issue_chunk(kbase, vbase, lk1, lv1, 1, tid);

    // prefix-combine segment summaries 0..j-1 (log2-domain exponents)
    float aa = 0.0f, bb = 0.0f, pp = -__builtin_inff();
    for (int js = 0; js < j; ++js) {
        const size_t o = ((size_t)b * S_SEG + js) * H_DIM + h;
        const float A = sumA[o], B = sumB[o], P = sumP[o];
        const float pd = pp + Lw2;
        const float q  = fmaxf(pd, P);
        const float e1 = EXP2F(pd - q);
        const float e2 = EXP2F(P - q);
        aa = aa * e1 + A * e2;
        bb = bb * e1 + B * e2;
        pp = q;
    }

    for (int c = 0; c < NCH; ++c) {
        if (c + 1 < NCH) asm volatile("s_wait_asynccnt 16" ::: "memory");
        else             asm volatile("s_wait_asynccnt 0"  ::: "memory");
        __syncthreads();

        const float* pk = &lds_k[c & 1][tid];
        const float* pv = &lds_v[c & 1][tid];
        float*       op = obase + (size_t)c * CHUNK * H_DIM;
#pragma unroll 4
        for (int tl = 0; tl < CHUNK; ++tl) {
            const float k2 = pk[tl * HB] * LOG2E;
            const float vt = pv[tl * HB];
            // stabilized wkv output (log2 domain)
            const float uk = u2 + k2;
            const float me = fmaxf(pp, uk);
            const float wt = EXP2F(uk - me);
            const float sc = EXP2F(pp - me);
            op[tl * H_DIM] = __fdividef(aa * sc + wt * vt, bb * sc + wt);
            // stabilized state update
            const float pw  = pp + w2;
            const float me2 = fmaxf(pw, k2);
            const float s1  = EXP2F(pw - me2);
            const float s2  = EXP2F(k2 - me2);
            aa = s1 * aa + s2 * vt;
            bb = s1 * bb + s2;
            pp = me2;
        }
        __syncthreads();
        if (c + 2 < NCH)
            issue_chunk(kbase, vbase, (c & 1) ? lk1 : lk0, (c & 1) ? lv1 : lv0,
                        c + 2, tid);
    }
}

extern "C" void kernel_launch(void* const* d_in, const int* in_sizes, int n_in,
                              void* d_out, int out_size, void* d_ws, size_t ws_size,
                              hipStream_t stream) {
    const float* key   = (const float*)d_in[0];
    const float* value = (const float*)d_in[1];
    const float* td    = (const float*)d_in[2];
    const float* tf    = (const float*)d_in[3];
    float*       out   = (float*)d_out;

    const int B = in_sizes[0] / (T_DIM * H_DIM);   // 8
    const size_t N = (size_t)B * S_SEG * H_DIM;

    float* ws   = (float*)d_ws;
    float* sumA = ws + 0 * N;
    float* sumB = ws + 1 * N;
    float* sumP = ws + 2 * N;

    dim3 grid1(H_DIM / HB, S_SEG - 1, B);          // last segment's summary unused
    dim3 grid3(H_DIM / HB, S_SEG, B);

    wkv_phase1<<<grid1, HB, 0, stream>>>(key, value, td, sumA, sumB, sumP);
    wkv_phase3<<<grid3, HB, 0, stream>>>(key, value, td, tf,
                                         sumA, sumB, sumP, out);
}